// ProteinMPNN_CPUkNN_83915071029335
// MI455X (gfx1250) — compile-verified
//
#include <hip/hip_runtime.h>

// ProteinMPNN kNN, fused WMMA distance + register-resident per-lane top-k.
// B=4, L=4096, K=48 (fixed by the reference).
//
// Per key tile: one v_wmma_f32_16x16x4_f32 produces the 16x16 D^2 tile
// directly (A row m = (-2kx,-2ky,-2kz,|k|^2), B col n = (qx,qy,qz,1),
// C = |q|^2). Selection runs in the squared-distance domain (sqrt is
// monotonic on clamped D^2); each lane keeps its top-48 as a sorted array of
// packed (d2_bits<<32 | idx) u64s in VGPRs, updated by a fully-unrolled
// v_min_u64/v_max_u64 ripple network guarded by an O(1) reject against the
// cached 48th best. The key-table ds_load_b128 is software-pipelined one
// tile ahead so LDS latency hides behind the candidate processing.
//
// d_out layout: [B*L*48] D_neighbors (float), then [B*L*48] E_idx (as float).

typedef unsigned long long ull;
typedef __attribute__((ext_vector_type(2))) float v2f;
typedef __attribute__((ext_vector_type(8))) float v8f;

#define Bn   4
#define Ln   4096
#define KNN  48
#define NTILE (Ln / 16)
#define BIGF 1000000000.0f

__global__ __launch_bounds__(32)
void knn_wmma16x16x4_kernel(const float* __restrict__ X,
                            const float* __restrict__ mask,
                            float* __restrict__ outD,
                            float* __restrict__ outE)
{
    extern __shared__ char smemraw[];
    float4* keytab   = (float4*)smemraw;                                   // 4096 * 16 B
    ull*    lists    = (ull*)(smemraw + (size_t)Ln * sizeof(float4));      // 32 * 48 * 8 B
    int*    s_nvalid = (int*)(smemraw + (size_t)Ln * sizeof(float4)
                                      + (size_t)32 * KNN * sizeof(ull));

    const int lane  = threadIdx.x;   // 0..31 (wave32)
    const int half  = lane >> 4;     // 0: K-dims 0,1   1: K-dims 2,3
    const int l16   = lane & 15;
    const int b     = blockIdx.y;
    const int qbase = blockIdx.x * 16;

    if (lane == 0) *s_nvalid = 0;
    __syncthreads();

    // ---- Build key table: entry = (-2x,-2y,-2z, |k|^2); masked -> (0,0,0, 2*BIG)
    int cnt = 0;
    for (int k = lane; k < Ln; k += 32) {
        const float* xp = X + (((size_t)(b * Ln + k)) * 4 + 1) * 3;  // Ca atom (index 1)
        float x = xp[0], y = xp[1], z = xp[2];
        float mv = mask[b * Ln + k];
        float4 e;
        if (mv > 0.0f) {
            e.x = -2.0f * x; e.y = -2.0f * y; e.z = -2.0f * z;
            e.w = x * x + y * y + z * z;
            cnt++;
        } else {
            e.x = 0.0f; e.y = 0.0f; e.z = 0.0f; e.w = 2.0f * BIGF;
        }
        keytab[k] = e;
    }
    atomicAdd(s_nvalid, cnt);

    // ---- Per-lane sorted top-48 list in VGPRs (ascending), sentinel = max
    ull lst[KNN];
    #pragma unroll
    for (int i = 0; i < KNN; ++i) lst[i] = ~0ull;
    ull worst = ~0ull;

    // ---- Query operands (wave-resident across the whole key sweep)
    const int q = qbase + l16;
    const float* xq = X + (((size_t)(b * Ln + q)) * 4 + 1) * 3;
    float qx = xq[0], qy = xq[1], qz = xq[2];
    float q2 = qx * qx + qy * qy + qz * qz;
    v2f bvec;                                   // B matrix 4x16: col n = (qx,qy,qz,1)
    bvec.x = half ? qz : qx;
    bvec.y = half ? 1.0f : qy;
    v8f cvec = { q2, q2, q2, q2, q2, q2, q2, q2 };  // C = |q|^2 broadcast

    __syncthreads();

    // ---- Sweep 256 key tiles, A-operand load pipelined one tile ahead
    float4 e = keytab[l16];                     // tile 0
    for (int t = 0; t < NTILE; ++t) {
        // Issue next tile's LDS load before processing this tile: its latency
        // hides behind the wmma + pack/compare/ripple work below.
        float4 enext = e;
        if (t + 1 < NTILE) enext = keytab[(t + 1) * 16 + l16];

        v2f a;                                  // A matrix 16x4: row m = keytab[m]
        a.x = half ? e.z : e.x;
        a.y = half ? e.w : e.y;
        v8f d2 = __builtin_amdgcn_wmma_f32_16x16x4_f32(
            false, a, false, bvec, (short)0, cvec, false, false);

        // lane l  (<16): keys 16t+0..7 of query l ; lane l+16: keys 16t+8..15
        const int base = t * 16 + (half ? 8 : 0);
        #pragma unroll
        for (int j = 0; j < 8; ++j) {
            float dv = fmaxf(d2[j], 0.0f);      // clamp: non-neg => bits monotonic
            ull pk = (((ull)__float_as_uint(dv)) << 32) | (unsigned)(base + j);
            if (pk < worst) {                    // O(1) reject vs cached 48th-best
                // Branchless ripple insert: bubble pk through the sorted list
                // with native v_min_u64/v_max_u64, evicting the old maximum.
                ull cur = pk;
                #pragma unroll
                for (int i = 0; i < KNN; ++i) {
                    ull lo = (cur < lst[i]) ? cur : lst[i];
                    ull hi = (cur < lst[i]) ? lst[i] : cur;
                    lst[i] = lo;
                    cur    = hi;
                }
                worst = lst[KNN - 1];
            }
        }
        e = enext;
    }

    // ---- Publish per-lane lists to LDS for the cross-lane merge
    ull* myOut = lists + (size_t)lane * KNN;
    #pragma unroll
    for (int i = 0; i < KNN; ++i) myOut[i] = lst[i];
    __syncthreads();

    // ---- Merge the two half-lists per query; sqrt only here (48/query);
    //      apply self/degenerate overrides
    const bool degen = (*s_nvalid <= KNN);
    if (lane < 16) {
        const int qq = qbase + lane;
        const bool self = degen || !(mask[b * Ln + qq] > 0.0f);
        const ull* La = lists + (size_t)lane * KNN;
        const ull* Lb = lists + (size_t)(lane + 16) * KNN;
        int ia = 0, ib = 0;
        const size_t obase = ((size_t)(b * Ln + qq)) * KNN;
        for (int j = 0; j < KNN; ++j) {
            ull va = La[ia], vb = Lb[ib];
            ull v;
            if (va <= vb) { v = va; ++ia; } else { v = vb; ++ib; }
            float dist = sqrtf(__uint_as_float((unsigned)(v >> 32)));
            int   idx  = (int)(v & 0xFFFFFFFFull);
            if (self) { dist = 0.0f; idx = qq; }
            outD[obase + j] = dist;
            outE[obase + j] = (float)idx;
        }
    }
}

extern "C" void kernel_launch(void* const* d_in, const int* in_sizes, int n_in,
                              void* d_out, int out_size, void* d_ws, size_t ws_size,
                              hipStream_t stream)
{
    (void)in_sizes; (void)n_in; (void)d_ws; (void)ws_size; (void)out_size;
    const float* X    = (const float*)d_in[0];   // (B,L,4,3) float32
    const float* mask = (const float*)d_in[1];   // (B,L)     float32
    float* outD = (float*)d_out;                         // (B,L,48) distances
    float* outE = outD + (size_t)Bn * Ln * KNN;          // (B,L,48) indices (as float)

    size_t smem = (size_t)Ln * sizeof(float4)            // key table   (64 KB)
                + (size_t)32 * KNN * sizeof(ull)         // merge lists (12 KB)
                + 16;                                    // n_valid counter
    dim3 grid(Ln / 16, Bn);                              // 256 query tiles x 4 batches
    knn_wmma16x16x4_kernel<<<grid, 32, smem, stream>>>(X, mask, outD, outE);
}